// PNAnet6L_80264348827995
// MI455X (gfx1250) — compile-verified
//
#include <hip/hip_runtime.h>
#include <cstdint>

// ---------------------------------------------------------------------------
// PNA 6-layer GNN forward for gfx1250 (MI455X).  fp32 throughout, WMMA
// V_WMMA_F32_16X16X4_F32 for every GEMM, scatter aggregation via L2 atomics,
// async global->LDS (ASYNCcnt) staging of the gathered edge operand.
// Scatter/post kernels are templated on tile sizes so accumulator tiles stay
// in statically-indexed VGPRs (no v_movrel indexed-VGPR traffic).
//
// Input layout assumption (setup_inputs() dict insertion order, nested dicts
// flattened in insertion order; n_in == 100):
//   0:x 1:edge_index 2:edge_attr 3:intarna_energy 4:batch 5:covalent_edges
//   6:dropout_conv_1_2 7:dropout_conv_rest
//   8..55:  convs[0..5] x {edge_w,edge_b,pre_w,pre_b,post_w,post_b,lin_w,lin_b}
//   56..65: bns[0..4] x {gamma,beta}
//   66..89: s2s[0..5] x {w_ih,w_hh,b_ih,b_hh}
//   90..99: head {w1,b1,g1,be1,w2,b2,w3,b3,w4,b4}
// Workspace requirement: ~260 MB.
// ---------------------------------------------------------------------------

#define NN 50000
#define NE 400000
#define NB 10
#define ADL 2.19722457733622f   /* log(9) */

typedef float v2f __attribute__((ext_vector_type(2)));
typedef float v8f __attribute__((ext_vector_type(8)));

// D(16x16,f32) = A(16x4,f32) * B(4x16,f32) + C  -- one wave, EXEC all ones.
__device__ __forceinline__ v8f wmma4(v2f a, v2f b, v8f c) {
#if defined(__gfx1250__)
  return __builtin_amdgcn_wmma_f32_16x16x4_f32(false, a, false, b, (short)0, c,
                                               false, false);
#else
  c[0] += a.x * b.x;  // host-pass placeholder, never executed
  return c;
#endif
}

// order-preserving float<->uint key (for atomicMin/Max on floats)
__device__ __forceinline__ unsigned fkey(float f) {
  unsigned u = __float_as_uint(f);
  return (u & 0x80000000u) ? ~u : (u | 0x80000000u);
}
__device__ __forceinline__ float dkey(unsigned k) {
  unsigned u = (k & 0x80000000u) ? (k & 0x7fffffffu) : ~k;
  return __uint_as_float(u);
}
__device__ __forceinline__ float sigm(float x) { return 1.f / (1.f + expf(-x)); }

// ------------------------------- utility -----------------------------------
__global__ void k_fill_f32(float* p, float v, int n) {
  int i = blockIdx.x * blockDim.x + threadIdx.x;
  if (i < n) p[i] = v;
}
__global__ void k_fill_u32(unsigned* p, unsigned v, int n) {
  int i = blockIdx.x * blockDim.x + threadIdx.x;
  if (i < n) p[i] = v;
}
__global__ void k_count(const int* __restrict__ dstI, unsigned* __restrict__ cnt) {
  int e = blockIdx.x * blockDim.x + threadIdx.x;
  if (e < NE) atomicAdd(&cnt[dstI[e]], 1u);
}
__global__ void k_scalers(const unsigned* __restrict__ cnt, float* __restrict__ scal) {
  int n = blockIdx.x * blockDim.x + threadIdx.x;
  if (n >= NN) return;
  unsigned c = cnt[n];
  float d = c > 0 ? (float)c : 1.f;
  float lg = logf(d + 1.f);
  scal[n * 3 + 0] = lg / ADL;   // amplification
  scal[n * 3 + 1] = ADL / lg;   // attenuation
  scal[n * 3 + 2] = d / 8.f;    // linear
}

// e = edge_attr(E,4) @ edge_w(4,F) + edge_b
__global__ void k_edge_lin(const float* __restrict__ ea, const float* __restrict__ w,
                           const float* __restrict__ b, float* __restrict__ out, int F) {
  int i = blockIdx.x * blockDim.x + threadIdx.x;
  if (i >= NE * F) return;
  int e = i / F, f = i - e * F;
  const float* a = ea + e * 4;
  float acc = b[f];
#pragma unroll
  for (int k = 0; k < 4; k++) acc += a[k] * w[k * F + f];
  out[i] = acc;
}

// -------- edge message GEMM + scatter aggregation (1 wave / 16 edges) ------
// h = [x_i | x_j | e] gathered straight into LDS via async loads (ASYNCcnt),
// then per K-step one LDS A-fragment feeds all N-tiles' WMMAs.  F is a
// template parameter so the acc tile array is statically indexed.
template <int F>
__global__ void __launch_bounds__(32)
k_scatter_t(const float* __restrict__ x, const float* __restrict__ ebuf,
            const int* __restrict__ srcI, const int* __restrict__ dstI,
            const float* __restrict__ pw, const float* __restrict__ pb,
            float* __restrict__ sumb, float* __restrict__ sqb,
            unsigned* __restrict__ mnk, unsigned* __restrict__ mxk) {
  constexpr int CIN = 2 * F;
  constexpr int K3 = 3 * F;
  constexpr int NTS = F / 16;
  __shared__ float Hs[2 * 16 * K3];
  int lane = threadIdx.x, l15 = lane & 15, half = lane >> 4, koff = half * 2;
  int e0 = blockIdx.x * 16;
#if defined(__gfx1250__)
  {
    // 16-byte chunks: 2 towers * 16 rows * 3 segments * (F/4) chunks.
    constexpr int CH = F / 4;
    constexpr int total = 96 * CH;  // multiple of 32 -> no EXEC divergence
    for (int idx = lane; idx < total; idx += 32) {
      int t = idx / (48 * CH);
      int rem = idx - t * 48 * CH;
      int r = rem / (3 * CH);
      int rem2 = rem - r * 3 * CH;
      int seg = rem2 / CH;
      int c = rem2 - seg * CH;
      int e = e0 + r;
      const float* gsrc;
      if (seg == 0)      gsrc = x + (size_t)dstI[e] * CIN + t * F + c * 4;
      else if (seg == 1) gsrc = x + (size_t)srcI[e] * CIN + t * F + c * 4;
      else               gsrc = ebuf + (size_t)e * F + c * 4;
      // flat->LDS mapping keeps the wave-relative LDS address in addr[31:0]
      unsigned ldsoff = (unsigned)(size_t)&Hs[(t * 16 + r) * K3 + seg * F + c * 4];
      asm volatile("global_load_async_to_lds_b128 %0, %1, off"
                   :: "v"(ldsoff), "v"(gsrc) : "memory");
    }
    asm volatile("s_wait_asynccnt 0x0" ::: "memory");
  }
#else
  for (int idx = lane; idx < 2 * 16 * K3; idx += 32) {
    int t = idx / (16 * K3);
    int rem = idx - t * 16 * K3;
    int r = rem / K3, k = rem - r * K3;
    int e = e0 + r;
    float val;
    if (k < F)            val = x[dstI[e] * CIN + t * F + k];
    else if (k < 2 * F)   val = x[srcI[e] * CIN + t * F + (k - F)];
    else                  val = ebuf[e * F + (k - 2 * F)];
    Hs[idx] = val;
  }
#endif
  __syncthreads();
  for (int t = 0; t < 2; t++) {
    v8f acc[NTS];
#pragma unroll
    for (int q = 0; q < NTS; q++) acc[q] = (v8f){};
    for (int k = 0; k < K3; k += 4) {
      v2f a;                                    // A loaded once per K-step,
      a.x = Hs[(t * 16 + l15) * K3 + k + koff]; // reused by all N-tiles
      a.y = Hs[(t * 16 + l15) * K3 + k + koff + 1];
#pragma unroll
      for (int nt = 0; nt < NTS; nt++) {
        v2f b;
        b.x = pw[(t * K3 + k + koff) * F + nt * 16 + l15];
        b.y = pw[(t * K3 + k + koff + 1) * F + nt * 16 + l15];
        acc[nt] = wmma4(a, b, acc[nt]);
      }
    }
#pragma unroll
    for (int nt = 0; nt < NTS; nt++) {
      float bc = pb[t * F + nt * 16 + l15];
#pragma unroll
      for (int v = 0; v < 8; v++) {
        int e = e0 + v + half * 8;
        int node = dstI[e];
        int o = node * CIN + t * F + nt * 16 + l15;
        float val = acc[nt][v] + bc;
        atomicAdd(sumb + o, val);
        atomicAdd(sqb + o, val * val);
        atomicMin(mnk + o, fkey(val));
        atomicMax(mxk + o, fkey(val));
      }
    }
  }
}

// finalize aggregation: agg[n][t][4F] = {mean, min, max, std}
__global__ void k_agg(const float* __restrict__ sumb, const float* __restrict__ sqb,
                      const unsigned* __restrict__ mnk, const unsigned* __restrict__ mxk,
                      const unsigned* __restrict__ cnt, float* __restrict__ agg,
                      int cin, int F) {
  int gid = blockIdx.x * blockDim.x + threadIdx.x;
  if (gid >= NN * cin) return;
  int n = gid / cin, i = gid - n * cin;
  int t = i / F, f = i - t * F;
  unsigned c = cnt[n];
  float cc = c > 0 ? (float)c : 1.f;
  float mean = sumb[gid] / cc;
  float msq = sqb[gid] / cc;
  float sd = sqrtf(fmaxf(msq - mean * mean, 0.f) + 1e-5f);
  float mn = c ? dkey(mnk[gid]) : 0.f;
  float mx = c ? dkey(mxk[gid]) : 0.f;
  float* o = agg + (n * 2 + t) * 4 * F;
  o[f] = mean; o[F + f] = mn; o[2 * F + f] = mx; o[3 * F + f] = sd;
}

// --- post MLP: [N,2,17F]@[2,17F,Fo]; A built on the fly into LDS (K-chunked)
template <int F, int Fo>
__global__ void __launch_bounds__(32)
k_post_t(const float* __restrict__ x, const float* __restrict__ agg,
         const float* __restrict__ scal, const float* __restrict__ pw,
         const float* __restrict__ pb, float* __restrict__ y) {
  constexpr int CIN = 2 * F;
  constexpr int K17 = 17 * F;
  constexpr int F4 = 4 * F;
  constexpr int NTS = Fo / 16;
  constexpr int KCH = (K17 < 544) ? K17 : 544;
  __shared__ float At[16 * KCH];
  int lane = threadIdx.x, l15 = lane & 15, half = lane >> 4, koff = half * 2;
  int m0 = blockIdx.x * 16;
  for (int t = 0; t < 2; t++) {
    v8f acc[NTS];
#pragma unroll
    for (int q = 0; q < NTS; q++) acc[q] = (v8f){};
    for (int kc = 0; kc < K17; kc += KCH) {
      int chunk = K17 - kc; if (chunk > KCH) chunk = KCH;
      for (int idx = lane; idx < 16 * chunk; idx += 32) {
        int r = idx / chunk, k = kc + (idx - r * chunk);
        int n = m0 + r;
        float val;
        if (k < F) val = x[n * CIN + t * F + k];            // x_t slice
        else {
          int j = k - F;
          int blk = j / F4;                                  // 0..3 scaler block
          int i4 = j - blk * F4;
          val = agg[(n * 2 + t) * F4 + i4];
          if (blk) val *= scal[n * 3 + blk - 1];
        }
        At[idx] = val;
      }
      __syncthreads();
      for (int kk = 0; kk < chunk; kk += 4) {
        v2f a;                                  // A once per K-step, all N-tiles
        a.x = At[l15 * chunk + kk + koff];
        a.y = At[l15 * chunk + kk + koff + 1];
        int kg = kc + kk + koff;
#pragma unroll
        for (int nt = 0; nt < NTS; nt++) {
          v2f b;
          b.x = pw[(t * K17 + kg) * Fo + nt * 16 + l15];
          b.y = pw[(t * K17 + kg + 1) * Fo + nt * 16 + l15];
          acc[nt] = wmma4(a, b, acc[nt]);
        }
      }
      __syncthreads();
    }
#pragma unroll
    for (int nt = 0; nt < NTS; nt++) {
      float bc = pb[t * Fo + nt * 16 + l15];
#pragma unroll
      for (int v = 0; v < 8; v++) {
        int row = m0 + v + half * 8;
        y[row * (2 * Fo) + t * Fo + nt * 16 + l15] = acc[nt][v] + bc;
      }
    }
  }
}

// generic fp32 WMMA GEMM: C[M,Nc] = A[M,K]@W[K,Nc] + bias (M clamped/masked)
__global__ void __launch_bounds__(32)
k_gemm(const float* __restrict__ A, const float* __restrict__ W,
       const float* __restrict__ bias, float* __restrict__ C,
       int M, int K, int Nc, int lda, int ldc, int relu) {
  int lane = threadIdx.x, l15 = lane & 15, half = lane >> 4, koff = half * 2;
  int m0 = blockIdx.x * 16, n0 = blockIdx.y * 16;
  int arow = m0 + l15; if (arow > M - 1) arow = M - 1;
  v8f acc = {};
  for (int k = 0; k < K; k += 4) {
    v2f a, b;
    a.x = A[arow * lda + k + koff];
    a.y = A[arow * lda + k + koff + 1];
    b.x = W[(k + koff) * Nc + n0 + l15];
    b.y = W[(k + koff + 1) * Nc + n0 + l15];
    acc = wmma4(a, b, acc);
  }
  float bc = bias ? bias[n0 + l15] : 0.f;
#pragma unroll
  for (int v = 0; v < 8; v++) {
    int row = m0 + v + half * 8;
    if (row < M) {
      float val = acc[v] + bc;
      if (relu) val = fmaxf(val, 0.f);
      C[row * ldc + n0 + l15] = val;
    }
  }
}

// ----------------------------- batch norm ----------------------------------
__global__ void k_bnstats(const float* __restrict__ x, int C, float* __restrict__ stat) {
  __shared__ float ss[256], sq[256];
  int c = blockIdx.x, tid = threadIdx.x;
  float s = 0.f, q = 0.f;
  for (int n = tid; n < NN; n += 256) { float v = x[n * C + c]; s += v; q += v * v; }
  ss[tid] = s; sq[tid] = q;
  __syncthreads();
  for (int o = 128; o > 0; o >>= 1) {
    if (tid < o) { ss[tid] += ss[tid + o]; sq[tid] += sq[tid + o]; }
    __syncthreads();
  }
  if (tid == 0) {
    float mean = ss[0] / NN;
    float var = sq[0] / NN - mean * mean;
    stat[2 * c] = mean;
    stat[2 * c + 1] = rsqrtf(var + 1e-5f);
  }
}
__global__ void k_bnapply(float* __restrict__ x, const float* __restrict__ stat,
                          const float* __restrict__ g, const float* __restrict__ bt,
                          int C, int do_bn) {
  int i = blockIdx.x * blockDim.x + threadIdx.x;
  if (i >= NN * C) return;
  int c = i % C;
  float v = x[i];
  if (do_bn) v = (v - stat[2 * c]) * stat[2 * c + 1] * g[c] + bt[c];
  x[i] = fmaxf(v, 0.f);
}

// ------------------------------ Set2Set ------------------------------------
// q is independent of the data: zero-init LSTM state => gates = b_ih + b_hh.
__global__ void k_s2s_q(const float* __restrict__ b_ih, const float* __restrict__ b_hh,
                        int C, float* __restrict__ qbuf, float* __restrict__ pooled,
                        int off) {
  int c = threadIdx.x;
  if (c >= C) return;
  float gi = b_ih[c] + b_hh[c];
  float gg = b_ih[2 * C + c] + b_hh[2 * C + c];
  float go = b_ih[3 * C + c] + b_hh[3 * C + c];
  float cc = sigm(gi) * tanhf(gg);   // f-gate * c0 term vanishes (c0 = 0)
  float q = sigm(go) * tanhf(cc);
  qbuf[c] = q;
  for (int b = 0; b < NB; b++) pooled[b * 896 + off + c] = q;
}
__global__ void k_s2s_e(const float* __restrict__ x, const float* __restrict__ q,
                        const int* __restrict__ batch, int C,
                        float* __restrict__ en, unsigned* __restrict__ emaxk) {
  int n = blockIdx.x * blockDim.x + threadIdx.x;
  if (n >= NN) return;
  float s = 0.f;
  for (int c = 0; c < C; c++) s += x[n * C + c] * q[c];
  en[n] = s;
  atomicMax(&emaxk[batch[n]], fkey(s));
}
__global__ void k_s2s_ex(float* __restrict__ en, const int* __restrict__ batch,
                         const unsigned* __restrict__ emaxk, float* __restrict__ denom) {
  int n = blockIdx.x * blockDim.x + threadIdx.x;
  if (n >= NN) return;
  int b = batch[n];
  float ex = expf(en[n] - dkey(emaxk[b]));
  en[n] = ex;
  atomicAdd(&denom[b], ex);
}
__global__ void k_s2s_r(const float* __restrict__ x, const float* __restrict__ en,
                        const float* __restrict__ denom, const int* __restrict__ batch,
                        int C, float* __restrict__ pooled, int off) {
  __shared__ float accs[128];
  __shared__ int uni, bsh;
  int tid = threadIdx.x;
  int n0 = blockIdx.x * 256;
  int n = n0 + tid;
  if (tid == 0) {
    int nl = n0 + 255; if (nl > NN - 1) nl = NN - 1;
    int b0 = batch[n0], b1 = batch[nl];
    uni = (b0 == b1); bsh = b0;
  }
  __syncthreads();
  if (uni) {  // batch is sorted: the common case
    for (int c = tid; c < C; c += 256) accs[c] = 0.f;
    __syncthreads();
    if (n < NN) {
      float a = en[n] / denom[bsh];
      for (int c = 0; c < C; c++) atomicAdd(&accs[c], a * x[n * C + c]);
    }
    __syncthreads();
    for (int c = tid; c < C; c += 256)
      atomicAdd(&pooled[bsh * 896 + off + C + c], accs[c]);
  } else {
    if (n < NN) {
      int b = batch[n];
      float a = en[n] / denom[b];
      for (int c = 0; c < C; c++)
        atomicAdd(&pooled[b * 896 + off + C + c], a * x[n * C + c]);
    }
  }
}

// -------------------------------- head -------------------------------------
__global__ void k_bn_head(float* __restrict__ z, const float* __restrict__ g,
                          const float* __restrict__ bt, int M, int C) {
  int c = blockIdx.x * blockDim.x + threadIdx.x;
  if (c >= C) return;
  float s = 0.f, q = 0.f;
  for (int m = 0; m < M; m++) { float v = z[m * C + c]; s += v; q += v * v; }
  float mean = s / M;
  float r = rsqrtf(q / M - mean * mean + 1e-5f);
  for (int m = 0; m < M; m++) {
    float v = (z[m * C + c] - mean) * r * g[c] + bt[c];
    z[m * C + c] = fmaxf(v, 0.f);
  }
}
__global__ void k_gemm_small(const float* __restrict__ A, const float* __restrict__ W,
                             const float* __restrict__ b, float* __restrict__ C_,
                             int M, int K, int Nc) {
  int i = blockIdx.x * blockDim.x + threadIdx.x;
  if (i >= M * Nc) return;
  int m = i / Nc, j = i - m * Nc;
  float s = b[j];
  for (int k = 0; k < K; k++) s += A[m * K + k] * W[k * Nc + j];
  C_[i] = s;
}

// ---------------------------------------------------------------------------
extern "C" void kernel_launch(void* const* d_in, const int* in_sizes, int n_in,
                              void* d_out, int out_size, void* d_ws, size_t ws_size,
                              hipStream_t stream) {
  (void)in_sizes; (void)n_in; (void)out_size; (void)ws_size;
  const float* x_in = (const float*)d_in[0];
  const int* ei = (const int*)d_in[1];
  const int* srcI = ei;
  const int* dstI = ei + NE;
  const float* edge_attr = (const float*)d_in[2];
  const int* batch = (const int*)d_in[4];

  int pi = 8;
  struct ConvP { const float *edge_w,*edge_b,*pre_w,*pre_b,*post_w,*post_b,*lin_w,*lin_b; } cv[6];
  for (int l = 0; l < 6; l++) {
    cv[l].edge_w=(const float*)d_in[pi++]; cv[l].edge_b=(const float*)d_in[pi++];
    cv[l].pre_w =(const float*)d_in[pi++]; cv[l].pre_b =(const float*)d_in[pi++];
    cv[l].post_w=(const float*)d_in[pi++]; cv[l].post_b=(const float*)d_in[pi++];
    cv[l].lin_w =(const float*)d_in[pi++]; cv[l].lin_b =(const float*)d_in[pi++];
  }
  struct BnP { const float *gamma,*beta; } bnp[5];
  for (int l = 0; l < 5; l++) {
    bnp[l].gamma=(const float*)d_in[pi++]; bnp[l].beta=(const float*)d_in[pi++];
  }
  struct S2SP { const float *w_ih,*w_hh,*b_ih,*b_hh; } sp[6];
  for (int l = 0; l < 6; l++) {
    sp[l].w_ih=(const float*)d_in[pi++]; sp[l].w_hh=(const float*)d_in[pi++];
    sp[l].b_ih=(const float*)d_in[pi++]; sp[l].b_hh=(const float*)d_in[pi++];
  }
  const float* hw1=(const float*)d_in[pi++]; const float* hb1=(const float*)d_in[pi++];
  const float* hg1=(const float*)d_in[pi++]; const float* hbe1=(const float*)d_in[pi++];
  const float* hw2=(const float*)d_in[pi++]; const float* hb2=(const float*)d_in[pi++];
  const float* hw3=(const float*)d_in[pi++]; const float* hb3=(const float*)d_in[pi++];
  const float* hw4=(const float*)d_in[pi++]; const float* hb4=(const float*)d_in[pi++];

  // ---- workspace carve (aliases: agg==ebuf, y==sumb) ----
  char* wp = (char*)d_ws;
  auto carve = [&](size_t bytes) {
    char* p = wp; wp += (bytes + 255) & ~(size_t)255; return p;
  };
  float*    ebuf  = (float*)   carve((size_t)NE * 64 * 4);   // edge msgs / agg
  float*    aggb  = ebuf;                                    // alias (>= N*8F)
  float*    sumb  = (float*)   carve((size_t)NN * 128 * 4);
  float*    y     = sumb;                                    // alias (post out)
  float*    sqb   = (float*)   carve((size_t)NN * 128 * 4);
  unsigned* mnk   = (unsigned*)carve((size_t)NN * 128 * 4);
  unsigned* mxk   = (unsigned*)carve((size_t)NN * 128 * 4);
  float*    xb0   = (float*)   carve((size_t)NN * 128 * 4);
  float*    xb1   = (float*)   carve((size_t)NN * 128 * 4);
  unsigned* cnt   = (unsigned*)carve((size_t)NN * 4);
  float*    scal  = (float*)   carve((size_t)NN * 3 * 4);
  float*    pooled= (float*)   carve((size_t)NB * 896 * 4);
  float*    qbuf  = (float*)   carve(128 * 4);
  float*    enb   = (float*)   carve((size_t)NN * 4);
  unsigned* emaxk = (unsigned*)carve(16 * 4);
  float*    denom = (float*)   carve(16 * 4);
  float*    stat  = (float*)   carve(2 * 896 * 4);
  float*    h1    = (float*)   carve((size_t)NB * 896 * 4);
  float*    h2    = (float*)   carve((size_t)NB * 384 * 4);
  float*    h3    = (float*)   carve((size_t)NB * 64 * 4);

  auto G = [](int n) { return dim3((unsigned)((n + 255) / 256)); };

  // degree & scalers depend only on dst -> compute once
  k_fill_u32<<<G(NN), 256, 0, stream>>>(cnt, 0u, NN);
  k_count<<<G(NE), 256, 0, stream>>>(dstI, cnt);
  k_scalers<<<G(NN), 256, 0, stream>>>(cnt, scal);
  k_fill_f32<<<G(NB * 896), 256, 0, stream>>>(pooled, 0.f, NB * 896);

  static const int CIN[6]  = {32, 32, 64, 128, 128, 64};
  static const int COUT[6] = {32, 64, 128, 128, 64, 32};
  const float* xcur = x_in;
  float* xbuf[2] = {xb0, xb1};
  int off = 0;

  for (int l = 0; l < 6; l++) {
    int cin = CIN[l], cout = COUT[l], F = cin / 2, Fo = cout / 2;

    k_edge_lin<<<G(NE * F), 256, 0, stream>>>(edge_attr, cv[l].edge_w,
                                              cv[l].edge_b, ebuf, F);
    int na = NN * cin;
    k_fill_f32<<<G(na), 256, 0, stream>>>(sumb, 0.f, na);
    k_fill_f32<<<G(na), 256, 0, stream>>>(sqb, 0.f, na);
    k_fill_u32<<<G(na), 256, 0, stream>>>(mnk, 0xFFFFFFFFu, na);
    k_fill_u32<<<G(na), 256, 0, stream>>>(mxk, 0u, na);

    switch (F) {  // compile-time tile shapes -> statically indexed acc VGPRs
      case 16: k_scatter_t<16><<<NE / 16, 32, 0, stream>>>(
                   xcur, ebuf, srcI, dstI, cv[l].pre_w, cv[l].pre_b,
                   sumb, sqb, mnk, mxk); break;
      case 32: k_scatter_t<32><<<NE / 16, 32, 0, stream>>>(
                   xcur, ebuf, srcI, dstI, cv[l].pre_w, cv[l].pre_b,
                   sumb, sqb, mnk, mxk); break;
      default: k_scatter_t<64><<<NE / 16, 32, 0, stream>>>(
                   xcur, ebuf, srcI, dstI, cv[l].pre_w, cv[l].pre_b,
                   sumb, sqb, mnk, mxk); break;
    }
    k_agg<<<G(na), 256, 0, stream>>>(sumb, sqb, mnk, mxk, cnt, aggb, cin, F);

    switch (l) {  // per-layer (F,Fo) instantiations
      case 0: k_post_t<16, 16><<<NN / 16, 32, 0, stream>>>(
                  xcur, aggb, scal, cv[l].post_w, cv[l].post_b, y); break;
      case 1: k_post_t<16, 32><<<NN / 16, 32, 0, stream>>>(
                  xcur, aggb, scal, cv[l].post_w, cv[l].post_b, y); break;
      case 2: k_post_t<32, 64><<<NN / 16, 32, 0, stream>>>(
                  xcur, aggb, scal, cv[l].post_w, cv[l].post_b, y); break;
      case 3: k_post_t<64, 64><<<NN / 16, 32, 0, stream>>>(
                  xcur, aggb, scal, cv[l].post_w, cv[l].post_b, y); break;
      case 4: k_post_t<64, 32><<<NN / 16, 32, 0, stream>>>(
                  xcur, aggb, scal, cv[l].post_w, cv[l].post_b, y); break;
      default: k_post_t<32, 16><<<NN / 16, 32, 0, stream>>>(
                  xcur, aggb, scal, cv[l].post_w, cv[l].post_b, y); break;
    }

    float* xn = xbuf[l & 1];
    k_gemm<<<dim3(NN / 16, cout / 16), 32, 0, stream>>>(
        y, cv[l].lin_w, cv[l].lin_b, xn, NN, cout, cout, cout, cout, 0);

    if (l < 5) {
      k_bnstats<<<cout, 256, 0, stream>>>(xn, cout, stat);
      k_bnapply<<<G(NN * cout), 256, 0, stream>>>(xn, stat, bnp[l].gamma,
                                                  bnp[l].beta, cout, 1);
    } else {
      k_bnapply<<<G(NN * cout), 256, 0, stream>>>(xn, stat, nullptr, nullptr,
                                                  cout, 0);
    }

    k_s2s_q<<<1, cout, 0, stream>>>(sp[l].b_ih, sp[l].b_hh, cout, qbuf, pooled, off);
    k_fill_u32<<<1, 32, 0, stream>>>(emaxk, 0u, NB);
    k_fill_f32<<<1, 32, 0, stream>>>(denom, 0.f, NB);
    k_s2s_e<<<G(NN), 256, 0, stream>>>(xn, qbuf, batch, cout, enb, emaxk);
    k_s2s_ex<<<G(NN), 256, 0, stream>>>(enb, batch, emaxk, denom);
    k_s2s_r<<<G(NN), 256, 0, stream>>>(xn, enb, denom, batch, cout, pooled, off);

    off += 2 * cout;
    xcur = xn;
  }

  // head: [10,896] -> bn+relu -> 384 -> 64 -> 2
  k_gemm<<<dim3(1, 896 / 16), 32, 0, stream>>>(pooled, hw1, hb1, h1,
                                               NB, 896, 896, 896, 896, 0);
  k_bn_head<<<7, 128, 0, stream>>>(h1, hg1, hbe1, NB, 896);
  k_gemm<<<dim3(1, 384 / 16), 32, 0, stream>>>(h1, hw2, hb2, h2,
                                               NB, 896, 384, 896, 384, 1);
  k_gemm<<<dim3(1, 64 / 16), 32, 0, stream>>>(h2, hw3, hb3, h3,
                                              NB, 384, 64, 384, 64, 1);
  k_gemm_small<<<1, 32, 0, stream>>>(h3, hw4, hb4, (float*)d_out, NB, 64, 2);
}